// SLAYFeatures_48765058679045
// MI455X (gfx1250) — compile-verified
//
#include <hip/hip_runtime.h>

typedef __attribute__((ext_vector_type(2))) float v2f;
typedef __attribute__((ext_vector_type(4))) float v4f;
typedef __attribute__((ext_vector_type(8))) float v8f;

#define NH     16
#define ND     64
#define NM     32
#define NR     2
#define XCOLS  1024   // H*D
#define OCOLS  1024   // H*R*M
#define TROWS  16     // rows per wave tile
#define NWAVES 8
#define WSTR   68     // LDS stride (floats) for Wt rows  -> bank-conflict-free b64 frags
#define XSTR   68     // LDS stride (floats) for x tile rows

__global__ __launch_bounds__(256) void slay_features_kernel(
    const float* __restrict__ x, const float* __restrict__ omega,
    const float* __restrict__ qn, const float* __restrict__ qw,
    float* __restrict__ out)
{
    // LDS: Wt[64 n][WSTR] shared by block, then per-wave x tiles [16][XSTR]
    __shared__ float lds[64 * WSTR + NWAVES * TROWS * XSTR];
    float* Wt = lds;

    const int tid  = threadIdx.x;
    const int wave = tid >> 5;
    const int lane = tid & 31;
    const int rrow = lane & 15;        // row (A) / col-in-tile (B,C)
    const int half = lane >> 4;        // 0: K%4 in {0,1}; 1: K%4 in {2,3}
    const int koff = half * 2;

    float* xt = lds + 64 * WSTR + wave * (TROWS * XSTR);
    const int rowBase = blockIdx.x * (NWAVES * TROWS) + wave * TROWS;

    // Quadrature constants (uniform scalar loads, tiny)
    const float s_r[2]   = { qn[0], qn[1] };
    const float sq2s[2]  = { sqrtf(2.0f * fmaxf(qn[0], 0.0f)),
                             sqrtf(2.0f * fmaxf(qn[1], 0.0f)) };
    const float sqw[2]   = { sqrtf(fmaxf(qw[0], 0.0f)),
                             sqrtf(fmaxf(qw[1], 0.0f)) };
    const float inv_m    = 1.0f / (float)NM;   // inv_sqrt_m * inv_sqrt_m

    for (int h = 0; h < NH; ++h) {
        __syncthreads();   // protect Wt from previous iteration's readers

        // ---- Stage omega[h] transposed into LDS: Wt[n=r*32+m][d] ----
        // omega is tiny (256 KB) and reused by every block: keep it cacheable (RT).
        for (int r = 0; r < NR; ++r) {
            const float* src = omega + (size_t)(r * NH + h) * (ND * NM);
            for (int i = tid; i < ND * NM; i += 256) {   // coalesced reads
                int d = i >> 5, m = i & 31;
                Wt[(r * 32 + m) * WSTR + d] = src[i];
            }
        }
        __syncthreads();

        // ---- Stage per-wave 16x64 x tile (coalesced NT b128 loads) ----
        // x is streamed once: non-temporal so it doesn't evict omega from L2.
        {
            const float* xs = x + (size_t)rowBase * XCOLS + h * ND;
            #pragma unroll
            for (int it = 0; it < 8; ++it) {
                int f   = it * 32 + lane;       // float4 index in tile
                int row = f >> 4;
                int c4  = (f & 15) << 2;
                v4f v = __builtin_nontemporal_load(
                            (const v4f*)(xs + (size_t)row * XCOLS + c4));
                *(v4f*)(xt + row * XSTR + c4) = v;
            }
        }

        // ---- A fragments (f32 16x4 layout) + row norm ----
        v2f a[16];
        float ss = 0.0f;
        #pragma unroll
        for (int k = 0; k < 16; ++k) {
            v2f f = *(const v2f*)(xt + rrow * XSTR + 4 * k + koff);
            a[k] = f;
            ss += f.x * f.x + f.y * f.y;
        }
        ss += __shfl_xor(ss, 16, 32);           // combine K-halves
        // 1/max(sqrt(ss),1e-6) == rsqrt(max(ss,1e-12)): single v_rsq_f32,
        // avoids the rcp+Newton+div_fixup chain of a true fp32 divide.
        const float scale = rsqrtf(fmaxf(ss, 1e-12f));
        #pragma unroll
        for (int k = 0; k < 16; ++k) { a[k].x *= scale; a[k].y *= scale; }

        // ---- GEMM: C[16x64] = A[16x64] * W[64x64] via V_WMMA_F32_16X16X4_F32 ----
        v8f acc[4] = {};
        #pragma unroll
        for (int k = 0; k < 16; ++k) {
            #pragma unroll
            for (int nt = 0; nt < 4; ++nt) {
                v2f b = *(const v2f*)(Wt + (nt * 16 + rrow) * WSTR + 4 * k + koff);
                acc[nt] = __builtin_amdgcn_wmma_f32_16x16x4_f32(
                    false, a[k], false, b, (short)0, acc[nt], false, false);
            }
        }

        // ---- Epilogue: p^2 * exp(clip(p*sqrt(2s)-s,-10,10)) * sqrt(w) / M ----
        // Output is streamed once: non-temporal stores.
        #pragma unroll
        for (int nt = 0; nt < 4; ++nt) {
            const int r   = nt >> 1;           // col = nt*16+n, r = col/32
            const int col = h * 64 + nt * 16 + rrow;
            #pragma unroll
            for (int v = 0; v < 8; ++v) {
                float p   = acc[nt][v];
                int   row = rowBase + v + 8 * half;   // C layout: M = v + 8*(lane>=16)
                float ea  = fminf(fmaxf(p * sq2s[r] - s_r[r], -10.0f), 10.0f);
                float val = p * p * inv_m * __expf(ea) * sqw[r];
                __builtin_nontemporal_store(val, out + (size_t)row * OCOLS + col);
            }
        }
    }
}

extern "C" void kernel_launch(void* const* d_in, const int* in_sizes, int n_in,
                              void* d_out, int out_size, void* d_ws, size_t ws_size,
                              hipStream_t stream) {
    const float* x     = (const float*)d_in[0];   // [B, 1024] fp32
    const float* omega = (const float*)d_in[1];   // [2,16,64,32] fp32
    const float* qn    = (const float*)d_in[2];   // [2]
    const float* qw    = (const float*)d_in[3];   // [2]
    float* out = (float*)d_out;                   // [B, 1024] fp32

    const int B      = in_sizes[0] / XCOLS;       // 65536
    const int blocks = B / (NWAVES * TROWS);      // 512 blocks of 128 rows

    slay_features_kernel<<<blocks, 256, 0, stream>>>(x, omega, qn, qw, out);
}